// MCDropout_81681688035444
// MI455X (gfx1250) — compile-verified
//
#include <hip/hip_runtime.h>
#include <stdint.h>

// out[b, j] = in[b, j] * mask[j]   (BATCH=32768, LAYER=4096, fp32)
//
// Memory-bound streaming kernel: ~1 GiB of traffic, ~134 MFLOP.
// Roofline @ 23.3 TB/s HBM => ~46 us; WMMA (dense GEMM vs diag) would be a
// ~8000x FLOP pessimization for identical bytes, so the matrix path is
// intentionally not used. CDNA5-specific footprint: async global->LDS mask
// broadcast (ASYNCcnt) + s_wait_asynccnt, NT-hinted b128 streams.

#define LAYER   4096
#define COLS4   (LAYER / 4)     // 1024 float4 columns
#define BLOCK   256             // 8 wave32s
#define GRID_X  (COLS4 / BLOCK) // 4 column groups
#define GRID_Y  2048            // each block loops BATCH/GRID_Y = 16 rows

typedef float float4v __attribute__((ext_vector_type(4)));

__global__ __launch_bounds__(BLOCK)
void MCDropout_81681688035444_kernel(const float* __restrict__ in,
                                     const float* __restrict__ mask,
                                     float* __restrict__ out,
                                     unsigned rows)
{
    __shared__ float s_mask[LAYER]; // 16 KiB of the 320 KiB WGP LDS

    const unsigned tid = threadIdx.x;

    // ---- Stage mask into LDS with CDNA5 async copy (tracked by ASYNCcnt) ----
    // 1024 b128 transfers; 256 lanes x 4 iterations.
    for (unsigned i = tid; i < COLS4; i += BLOCK) {
        unsigned long long gaddr =
            (unsigned long long)(uintptr_t)(mask + (size_t)i * 4);
        unsigned lds_off = (unsigned)(uintptr_t)(&s_mask[i * 4]); // low 32b = LDS offset
        asm volatile("global_load_async_to_lds_b128 %0, %1, off"
                     :: "v"(lds_off), "v"(gaddr)
                     : "memory");
    }
    asm volatile("s_wait_asynccnt 0x0" ::: "memory"); // my async copies landed
    __syncthreads();                                  // all waves' copies visible

    // ---- Each thread owns one float4 column; mask lives in registers ----
    const unsigned c4 = blockIdx.x * BLOCK + tid;          // 0..1023
    const float4v m = *(const float4v*)(&s_mask[c4 * 4]);  // ds_load_b128, once

    const float4v* __restrict__ inv  = (const float4v*)in;
    float4v*       __restrict__ outv = (float4v*)out;

    // Stream rows: coalesced 16B/lane, non-temporal (each byte touched once;
    // 512 MiB stream > 192 MB L2). Unroll x4 so the compiler clauses 4
    // b128 NT loads back-to-back -> 4x memory-level parallelism per wave.
#pragma unroll 4
    for (unsigned r = blockIdx.y; r < rows; r += GRID_Y) {
        size_t idx = (size_t)r * COLS4 + c4;
        float4v v = __builtin_nontemporal_load(&inv[idx]);
        v *= m;
        __builtin_nontemporal_store(v, &outv[idx]);
    }
}

extern "C" void kernel_launch(void* const* d_in, const int* in_sizes, int n_in,
                              void* d_out, int out_size, void* d_ws, size_t ws_size,
                              hipStream_t stream)
{
    const float* in   = (const float*)d_in[0]; // (32768, 4096) f32
    const float* mask = (const float*)d_in[1]; // (4096,) f32
    float*       out  = (float*)d_out;

    const unsigned rows = (unsigned)(in_sizes[0] / LAYER); // 32768

    dim3 grid(GRID_X, GRID_Y, 1);
    dim3 block(BLOCK, 1, 1);
    MCDropout_81681688035444_kernel<<<grid, block, 0, stream>>>(in, mask, out, rows);
}